// MixHopConv_32117765439686
// MI455X (gfx1250) — compile-verified
//
#include <hip/hip_runtime.h>
#include <hip/hip_bf16.h>

// MixHop GCN on MI455X (gfx1250, wave32).
// Strategy:
//  - deg/norm via float atomics (all traffic L2-resident: h is 25.6MB << 192MB L2)
//  - fold final FC into per-hop weights: M_j = Wfc_j @ W_j  (done on device),
//    so out = bias_tot + sum_j h_j @ M_j^T  -> no [N,384] concat buffer
//  - SpMM scatter: 1 wave per edge, float4 loads, global_atomic_add_f32
//  - GEMM hops: fp32 WMMA (v_wmma_f32_16x16x4_f32), 16x128 stripe per block

typedef float v2f __attribute__((ext_vector_type(2)));
typedef float v8f __attribute__((ext_vector_type(8)));

#define DIN 128
#define DHOPS 3

// ---------------- degree / norm ----------------
__global__ void k_degree(const int* __restrict__ dst, float* __restrict__ deg, int E) {
    int e = blockIdx.x * blockDim.x + threadIdx.x;
    if (e < E) atomicAdd(&deg[dst[e]], 1.0f);
}

__global__ void k_norm(float* __restrict__ norm, int N) {
    int n = blockIdx.x * blockDim.x + threadIdx.x;
    if (n < N) {
        float d = norm[n];           // deg stored in-place
        if (d < 1.0f) d = 1.0f;
        norm[n] = rsqrtf(d);
    }
}

// ---------------- fold Wfc into per-hop weights ----------------
// Mall[j][o][i] = sum_k Wfc[o][j*128+k] * Wj[k][i]
__global__ void k_combine(const float* __restrict__ Wfc,
                          const float* __restrict__ W0,
                          const float* __restrict__ W1,
                          const float* __restrict__ W2,
                          float* __restrict__ Mall) {
    int t = blockIdx.x * blockDim.x + threadIdx.x;
    if (t >= DHOPS * DIN * DIN) return;
    int j = t >> 14;
    int o = (t >> 7) & 127;
    int i = t & 127;
    const float* Wj = (j == 0) ? W0 : ((j == 1) ? W1 : W2);
    const float* wf = Wfc + (size_t)o * (DHOPS * DIN) + j * DIN;
    float acc = 0.0f;
    #pragma unroll 8
    for (int k = 0; k < DIN; ++k) acc = fmaf(wf[k], Wj[(size_t)k * DIN + i], acc);
    Mall[t] = acc;
}

// bias_tot[o] = bfc[o] + sum_j sum_k Wfc[o][j*128+k] * bj[k]
__global__ void k_biasvec(const float* __restrict__ Wfc,
                          const float* __restrict__ b0, const float* __restrict__ b1,
                          const float* __restrict__ b2, const float* __restrict__ bfc,
                          float* __restrict__ bias) {
    int o = threadIdx.x;   // 128 threads
    const float* wf = Wfc + (size_t)o * (DHOPS * DIN);
    float acc = bfc[o];
    for (int k = 0; k < DIN; ++k) {
        acc = fmaf(wf[k],           b0[k], acc);
        acc = fmaf(wf[DIN + k],     b1[k], acc);
        acc = fmaf(wf[2 * DIN + k], b2[k], acc);
    }
    bias[o] = acc;
}

__global__ void k_bias_init(const float* __restrict__ bias, float* __restrict__ out, int N) {
    int t = blockIdx.x * blockDim.x + threadIdx.x;
    if (t < N * DIN) out[t] = bias[t & 127];
}

// ---------------- SpMM scatter: nxt[dst] += cur[src] * norm[src] ----------------
// one wave (32 lanes) per edge; each lane handles 4 contiguous floats
__global__ void k_scatter(const float* __restrict__ cur, const float* __restrict__ norm,
                          const int* __restrict__ src, const int* __restrict__ dst,
                          float* __restrict__ nxt, int E) {
    long long t = (long long)blockIdx.x * blockDim.x + threadIdx.x;
    int e = (int)(t >> 5);
    if (e >= E) return;
    int f = ((int)t & 31) << 2;
    int s = src[e];
    int d = dst[e];
    float ns = norm[s];
    const float4 v = *(const float4*)(cur + (size_t)s * DIN + f);
    float* p = nxt + (size_t)d * DIN + f;
    atomicAdd(p + 0, v.x * ns);
    atomicAdd(p + 1, v.y * ns);
    atomicAdd(p + 2, v.z * ns);
    atomicAdd(p + 3, v.w * ns);
}

// ---------------- post-scale: h[n][:] *= norm[n] ----------------
__global__ void k_scale(float* __restrict__ h, const float* __restrict__ norm, int N) {
    int t = blockIdx.x * blockDim.x + threadIdx.x;   // N*32 threads, float4 each
    if (t >= N * (DIN / 4)) return;
    int n = t >> 5;
    float s = norm[n];
    float4* p = (float4*)h + t;
    float4 v = *p;
    v.x *= s; v.y *= s; v.z *= s; v.w *= s;
    *p = v;
}

// ---------------- WMMA GEMM: out[16 rows x 128 cols] += h_tile @ M^T ----------------
// block = 256 threads = 8 waves; wave w computes the 16x16 tile at cols [16w,16w+16)
// fp32 WMMA 16x16x4: A lane layout: row = lane%16, K pair offset = 2*(lane/16)
//                    C/D: VGPR v -> row v + 8*(lane/16), col = lane%16
__global__ void k_gemm_wmma(const float* __restrict__ h, const float* __restrict__ M,
                            float* __restrict__ out, int N) {
    const int lane = threadIdx.x & 31;
    const int wave = threadIdx.x >> 5;       // 0..7 -> col tile
    const int n0   = blockIdx.x * 16;        // row tile (N divisible by 16)
    const int row  = lane & 15;
    const int col  = lane & 15;
    const int koff = (lane >> 4) << 1;       // 0 or 2
    const int rhalf = (lane >> 4) << 3;      // 0 or 8 (C/D row half)
    const int o0   = wave << 4;

    // load existing C tile (bias + previous hops' contributions)
    v8f c;
    #pragma unroll
    for (int v = 0; v < 8; ++v)
        c[v] = out[(size_t)(n0 + rhalf + v) * DIN + o0 + col];

    const float* hrow = h + (size_t)(n0 + row) * DIN;
    const float* mcol = M + (size_t)(o0 + col) * DIN;   // B = M^T: col of B == row of M

    #pragma unroll 4
    for (int k = 0; k < DIN; k += 4) {
        v2f a, b;
        a.x = hrow[k + koff];
        a.y = hrow[k + koff + 1];
        b.x = mcol[k + koff];
        b.y = mcol[k + koff + 1];
        c = __builtin_amdgcn_wmma_f32_16x16x4_f32(
                /*neg_a=*/false, a, /*neg_b=*/false, b,
                /*c_mod=*/(short)0, c, /*reuse_a=*/false, /*reuse_b=*/false);
    }

    #pragma unroll
    for (int v = 0; v < 8; ++v)
        out[(size_t)(n0 + rhalf + v) * DIN + o0 + col] = c[v];
}

// ---------------- host launch ----------------
extern "C" void kernel_launch(void* const* d_in, const int* in_sizes, int n_in,
                              void* d_out, int out_size, void* d_ws, size_t ws_size,
                              hipStream_t stream) {
    const float* feat = (const float*)d_in[0];
    const float* W0   = (const float*)d_in[1];
    const float* b0   = (const float*)d_in[2];
    const float* W1   = (const float*)d_in[3];
    const float* b1   = (const float*)d_in[4];
    const float* W2   = (const float*)d_in[5];
    const float* b2   = (const float*)d_in[6];
    const float* Wfc  = (const float*)d_in[7];
    const float* bfc  = (const float*)d_in[8];
    const int*   ei   = (const int*)d_in[9];   // [2, E] row-major

    const int N = in_sizes[0] / DIN;
    const int E = in_sizes[9] / 2;
    const int* src = ei;
    const int* dst = ei + E;

    float* out = (float*)d_out;

    // workspace carve-up (256B aligned)
    auto align256 = [](size_t x) { return (x + 255) & ~(size_t)255; };
    char* ws = (char*)d_ws;
    size_t off = 0;
    float* norm = (float*)(ws + off); off += align256((size_t)N * 4);
    float* Mall = (float*)(ws + off); off += align256((size_t)DHOPS * DIN * DIN * 4);
    float* bias = (float*)(ws + off); off += align256((size_t)DIN * 4);
    float* hA   = (float*)(ws + off); off += align256((size_t)N * DIN * 4);
    float* hB   = (float*)(ws + off); off += align256((size_t)N * DIN * 4);
    (void)ws_size; (void)n_in; (void)out_size;

    const int BT = 256;

    // degree -> norm
    hipMemsetAsync(norm, 0, (size_t)N * 4, stream);
    k_degree<<<(E + BT - 1) / BT, BT, 0, stream>>>(dst, norm, E);
    k_norm<<<(N + BT - 1) / BT, BT, 0, stream>>>(norm, N);

    // fold FC into hop weights; init output with combined bias
    k_combine<<<(DHOPS * DIN * DIN + BT - 1) / BT, BT, 0, stream>>>(Wfc, W0, W1, W2, Mall);
    k_biasvec<<<1, DIN, 0, stream>>>(Wfc, b0, b1, b2, bfc, bias);
    k_bias_init<<<((size_t)N * DIN + BT - 1) / BT, BT, 0, stream>>>(bias, out, N);

    // 3 propagation hops
    const float* cur = feat;
    float* bufs[2] = { hA, hB };
    long long scatter_threads = (long long)E * 32;
    int scatter_blocks = (int)((scatter_threads + BT - 1) / BT);
    int scale_blocks = (N * (DIN / 4) + BT - 1) / BT;
    int gemm_blocks = N / 16;   // N = 50000 divisible by 16

    for (int j = 0; j < DHOPS; ++j) {
        float* nxt = bufs[j & 1];
        hipMemsetAsync(nxt, 0, (size_t)N * DIN * 4, stream);
        k_scatter<<<scatter_blocks, BT, 0, stream>>>(cur, norm, src, dst, nxt, E);
        k_scale<<<scale_blocks, BT, 0, stream>>>(nxt, norm, N);
        k_gemm_wmma<<<gemm_blocks, BT, 0, stream>>>(nxt, Mall + (size_t)j * DIN * DIN, out, N);
        cur = nxt;
    }
}